// Graph2MapFusion_85134841741351
// MI455X (gfx1250) — compile-verified
//
#include <hip/hip_runtime.h>
#include <hip/hip_bf16.h>

typedef __attribute__((ext_vector_type(2))) float v2f;
typedef __attribute__((ext_vector_type(4))) float v4f;
typedef __attribute__((ext_vector_type(8))) float v8f;

#define BATCH 8
#define KNOD  100000
#define CCH   16
#define OCH   16
#define HG    512
#define WG    512
#define HWG   (HG * WG)
// workspace grid: [b][y][x][o] NHWC, 8*512*512*16 f32 = 134,217,728 B (L2-resident)
#define GRID_FLOATS ((size_t)BATCH * HWG * OCH)

// ---------------------------------------------------------------------------
// Kernel 1: zero the workspace grid (float4 stores, exactly covers the grid)
// ---------------------------------------------------------------------------
__global__ __launch_bounds__(256) void g2m_zero(float* __restrict__ ws) {
  size_t i = (size_t)blockIdx.x * blockDim.x + threadIdx.x;  // grid sized exactly
  ((v4f*)ws)[i] = (v4f)0.0f;
}

// ---------------------------------------------------------------------------
// Kernel 2: per-wave, take 16 nodes, channel-mix with W_obj/W_prior via
// V_WMMA_F32_16X16X4_F32 (8 chained WMMAs), then atomic-scatter the mixed
// 16-vectors into the NHWC grid.
// ---------------------------------------------------------------------------
__global__ __launch_bounds__(256) void g2m_scatter(
    const float* __restrict__ feat,   // [B,K,16]
    const float* __restrict__ xy,     // [B,K,2]
    const int*   __restrict__ types,  // [B,K]
    const float* __restrict__ w_obj,  // [16,16] row-major [O][C]
    const float* __restrict__ w_prior,
    float* __restrict__ grid)         // [B][H][W][16]
{
  const int lane = threadIdx.x & 31;
  const int wv   = blockIdx.x * (blockDim.x >> 5) + (threadIdx.x >> 5);
  const long g0  = (long)wv * 16;      // first flattened node (b*K + k) of group
  const int row  = lane & 15;          // node-in-group for A; out-channel for D
  const int half = lane >> 4;          // which K-half (A) / M-half (D)
  const long n   = g0 + row;           // node this lane loads
  const int  b   = (int)(g0 / KNOD);   // uniform: 100000 % 16 == 0

  // per-node pixel index (jnp.round == rintf, round-half-even)
  float fx = xy[n * 2 + 0];
  float fy = xy[n * 2 + 1];
  int xs = (int)fminf(fmaxf(rintf(fx), 0.0f), (float)(WG - 1));
  int ys = (int)fminf(fmaxf(rintf(fy), 0.0f), (float)(HG - 1));
  int idx = ys * WG + xs;

  int   t  = types[n];
  float mo = (t == 0) ? 1.0f : 0.0f;
  float mp = (t == 1) ? 1.0f : 0.0f;

  // D[m, o] = sum_c feat[m,c]*mo*Wobj[o,c] + feat[m,c]*mp*Wpri[o,c]
  // A 16x4 f32 layout: lanes0-15 K={kb,kb+1}, lanes16-31 K={kb+2,kb+3}
  const float* fb = feat + n * CCH;
  v8f acc = (v8f)0.0f;
#pragma unroll
  for (int j = 0; j < 4; ++j) {
    const int kb = 4 * j + 2 * half;
    v2f f  = *(const v2f*)(fb + kb);
    v2f ao = f * mo;
    v2f ap = f * mp;
    v2f bo = *(const v2f*)(w_obj   + row * CCH + kb);  // B[k,n] = W[n,k]
    v2f bp = *(const v2f*)(w_prior + row * CCH + kb);
    acc = __builtin_amdgcn_wmma_f32_16x16x4_f32(false, ao, false, bo,
                                                (short)0, acc, false, false);
    acc = __builtin_amdgcn_wmma_f32_16x16x4_f32(false, ap, false, bp,
                                                (short)0, acc, false, false);
  }

  // D layout: VGPR r -> node (r + 8*half), N (out-channel) = lane&15.
  // Fetch that node's pixel index from the lane that computed it.
  float* gb = grid + (size_t)b * ((size_t)HWG * OCH);
#pragma unroll
  for (int r = 0; r < 8; ++r) {
    int src  = r + 8 * half;            // lane owning node (g0+src)'s idx
    int nidx = __shfl(idx, src, 32);
    float* p = gb + (size_t)nidx * OCH + row;
    __hip_atomic_fetch_add(p, acc[r], __ATOMIC_RELAXED, __HIP_MEMORY_SCOPE_AGENT);
  }
}

// ---------------------------------------------------------------------------
// Kernel 3: depthwise 3x3 conv on the mixed grid, NHWC->NCHW output.
// 16x16 pixel tile + 1-pixel halo staged in LDS (pixel stride 20 dwords to
// avoid bank conflicts), one pixel (all 16 channels) per thread.
// ---------------------------------------------------------------------------
#define PS 20  // LDS floats per pixel (16 data + 4 pad), keeps 16B alignment
__global__ __launch_bounds__(256) void g2m_conv(const float* __restrict__ grid,
                                                float* __restrict__ out) {
  __shared__ float lds[18 * 18 * PS];
  const int b  = blockIdx.z;
  const int ty = blockIdx.y * 16;
  const int tx = blockIdx.x * 16;
  const float* gb = grid + (size_t)b * ((size_t)HWG * OCH);

  for (int p = threadIdx.x; p < 18 * 18; p += 256) {
    int py = p / 18, px = p % 18;
    int gy = ty + py - 1, gx = tx + px - 1;
    v4f v0 = 0.0f, v1 = 0.0f, v2 = 0.0f, v3 = 0.0f;
    if (gy >= 0 && gy < HG && gx >= 0 && gx < WG) {
      const v4f* s = (const v4f*)(gb + ((size_t)gy * WG + gx) * OCH);
      v0 = s[0]; v1 = s[1]; v2 = s[2]; v3 = s[3];
    }
    v4f* d = (v4f*)&lds[p * PS];
    d[0] = v0; d[1] = v1; d[2] = v2; d[3] = v3;
  }
  __syncthreads();

  const int y = threadIdx.x >> 4;
  const int x = threadIdx.x & 15;
  const float KW[3][3] = {{0.075f, 0.125f, 0.075f},
                          {0.125f, 0.300f, 0.125f},
                          {0.075f, 0.125f, 0.075f}};
  v4f a0 = 0.0f, a1 = 0.0f, a2 = 0.0f, a3 = 0.0f;
#pragma unroll
  for (int dy = 0; dy < 3; ++dy) {
#pragma unroll
    for (int dx = 0; dx < 3; ++dx) {
      const float w = KW[dy][dx];
      const v4f* s = (const v4f*)&lds[((y + dy) * 18 + (x + dx)) * PS];
      a0 += w * s[0]; a1 += w * s[1]; a2 += w * s[2]; a3 += w * s[3];
    }
  }

  const int gy = ty + y, gx = tx + x;
  float av[16];
  *(v4f*)&av[0] = a0; *(v4f*)&av[4]  = a1;
  *(v4f*)&av[8] = a2; *(v4f*)&av[12] = a3;
  // out[b][o][gy][gx]; lanes 0-15 share gy, consecutive gx -> coalesced per o
  float* ob = out + (size_t)b * OCH * HWG + (size_t)gy * WG + gx;
#pragma unroll
  for (int o = 0; o < 16; ++o) ob[(size_t)o * HWG] = av[o];
}

// ---------------------------------------------------------------------------
extern "C" void kernel_launch(void* const* d_in, const int* in_sizes, int n_in,
                              void* d_out, int out_size, void* d_ws, size_t ws_size,
                              hipStream_t stream) {
  const float* node_feat  = (const float*)d_in[0];  // [8,100000,16] f32
  const float* node_xy    = (const float*)d_in[1];  // [8,100000,2]  f32
  // d_in[2] = hw (int64[2]) -- constants 512x512 baked in
  const int*   node_types = (const int*)d_in[3];    // [8,100000]    i32
  const float* w_obj      = (const float*)d_in[4];  // [16,16]       f32
  const float* w_prior    = (const float*)d_in[5];  // [16,16]       f32
  float* out  = (float*)d_out;                      // [8,16,512,512] f32
  float* grid = (float*)d_ws;                       // GRID_FLOATS f32

  // 1) zero grid: GRID_FLOATS/4 float4 stores, 256 threads/block
  const int zero_blocks = (int)(GRID_FLOATS / 4 / 256);   // 32768, exact
  g2m_zero<<<zero_blocks, 256, 0, stream>>>(grid);

  // 2) WMMA channel-mix + atomic scatter: 800000 nodes / 16-per-wave / 8 waves
  const int scat_blocks = (BATCH * KNOD) / 16 / 8;        // 6250, exact
  g2m_scatter<<<scat_blocks, 256, 0, stream>>>(node_feat, node_xy, node_types,
                                               w_obj, w_prior, grid);

  // 3) conv + transpose to NCHW output
  dim3 cgrid(WG / 16, HG / 16, BATCH);                    // 32 x 32 x 8
  g2m_conv<<<cgrid, 256, 0, stream>>>(grid, out);
}